// SegDiceLoss_40407052321225
// MI455X (gfx1250) — compile-verified
//
#include <hip/hip_runtime.h>

// SegDiceLoss on MI455X (gfx1250, wave32).
// Per (b,n): gather 169-float MLP params, run 10->8->8->1 MLP over 16384 pixels
// using V_WMMA_F32_16X16X4_F32 chains (exact fp32), then dice-loss reduction.
//
// Duplicated-weight-row trick: A-matrix rows 8..15 replicate rows 0..7, so the
// 16x16 D accumulator holds each h-row duplicated across both lane halves and
// the next WMMA's B operand is built with plain v_cndmask (no lane permutes).
//
// Hot-loop optimizations this round:
//  * no divergent branch: both lane halves accumulate identical dice partials
//    (same pixel, same cacheline), result scaled by 0.5
//  * x_rel hoisted (loop-invariant); y_rel strength-reduced to one v_add with
//    the hi-lane zeroing folded into the initial value & step (no cndmask)
//  * all seg_feat loads share one 32-bit voffset (channel rows addressed via
//    the 24-bit instruction offset), one v_add_nc_u32 advance per iteration

typedef float v2f __attribute__((ext_vector_type(2)));
typedef float v8f __attribute__((ext_vector_type(8)));

namespace {
constexpr int kB  = 8;
constexpr int kN  = 64;
constexpr int kC  = 8;
constexpr int kW  = 128;
constexpr int kHW = 128 * 128;   // 16384 pixels
constexpr int kD  = 169;         // (C+2)*C + C + C*C + C + C + 1
// parameter-vector offsets
constexpr int kW1 = 0;    // 8 x 10
constexpr int kB1 = 80;   // 8
constexpr int kW2 = 88;   // 8 x 8
constexpr int kB2 = 152;  // 8
constexpr int kW3 = 160;  // 8
constexpr int kB3 = 168;  // 1
}

__device__ __forceinline__ v8f wmma_f32_k4(v2f a, v2f b, v8f c) {
  // D = A(16x4,f32) * B(4x16,f32) + C(16x16,f32); exact fp32 matrix op.
  return __builtin_amdgcn_wmma_f32_16x16x4_f32(
      /*neg_a=*/false, a, /*neg_b=*/false, b,
      /*c_mod=*/(short)0, c, /*reuse_a=*/false, /*reuse_b=*/false);
}

__global__ __launch_bounds__(256) void seg_mlp_partial(
    const float* __restrict__ seg_feat,     // (8, 8, 128, 128)
    const float* __restrict__ conv_weight,  // (8, 169, 128, 128)
    const float* __restrict__ mask,         // (8, 64)
    const int*   __restrict__ ind,          // (8, 64)  (harness passes int32)
    const float* __restrict__ target,       // (8, 64, 128, 128)
    float* __restrict__ partial)            // (512, 3): inter, pred2, tgt2
{
  __shared__ float w[kD];
  __shared__ float red[8][3];

  const int blk  = blockIdx.x;          // 0..511
  const int b    = blk >> 6;
  const int n    = blk & 63;
  const int tid  = threadIdx.x;
  const int lane = tid & 31;
  const int wv   = tid >> 5;            // wave 0..7

  const int p0 = ind[b * kN + n];       // gather pixel for this point
  if (tid < kD)
    w[tid] = conv_weight[((size_t)b * kD + tid) * kHW + p0];
  __syncthreads();

  const float m  = mask[b * kN + n];
  const float x0 = (float)(p0 & (kW - 1));
  const float y0 = (float)(p0 >> 7);

  const int mrow = lane & 15;
  const int m8   = mrow & 7;            // duplicated output row
  const int hi   = lane >> 4;           // 0: lanes 0-15, 1: lanes 16-31

  // ---- Preload A operands (weights, duplicated rows) into VGPRs ----
  v2f a1[3], a2[2];
#pragma unroll
  for (int s = 0; s < 3; ++s)
#pragma unroll
    for (int v = 0; v < 2; ++v) {
      const int k = 4 * s + hi * 2 + v;               // K index 0..11
      a1[s][v] = (k < 10) ? w[kW1 + m8 * 10 + k] : 0.f;
    }
#pragma unroll
  for (int s = 0; s < 2; ++s)
#pragma unroll
    for (int v = 0; v < 2; ++v) {
      const int k = 4 * s + hi * 2 + v;               // K index 0..7
      a2[s][v] = w[kW2 + m8 * 8 + k];
    }

  float bias1[8], bias2[8], w3v[8];
#pragma unroll
  for (int r = 0; r < 8; ++r) {
    bias1[r] = w[kB1 + r];
    bias2[r] = w[kB2 + r];
    w3v[r]   = w[kW3 + r];
  }
  const float b3 = w[kB3];

  // Uniform bases; per-lane 32-bit element offsets (fits easily in 32 bits).
  // seg_feat rows for GEMM1 B operand: step0 rows {0,2},{1,3}; step1 {4,6},{5,7}.
  const float* __restrict__ sfb = seg_feat + (size_t)b * kC * kHW;
  const float* __restrict__ tg  = target + ((size_t)b * kN + n) * kHW;
  unsigned voff  = (unsigned)((hi ? 2 : 0) * kHW + wv * 16 + mrow);
  unsigned tvoff = (unsigned)(wv * 16 + mrow);

  // x_rel is loop-invariant (pix & 127 == wv*16 + mrow); y_rel is an
  // arithmetic sequence. Fold hi-lane zeroing into value & step: hi lanes
  // start at 0 and add 0 forever -> no per-iteration cndmask.
  const float xr_v   = hi ? 0.f : ((float)(wv * 16 + mrow) - x0) * (1.f / 128.f);
  float       yr_v   = hi ? 0.f : (0.f - y0) * (1.f / 128.f);
  const float yr_stp = hi ? 0.f : (1.f / 128.f);

  float inter = 0.f, p2 = 0.f, t2 = 0.f;

#pragma unroll 2
  for (int it = 0; it < 128; ++it) {
    // ---- B operand: feature columns (channels 0..7, x_rel, y_rel, 0, 0) ----
    v2f bs0, bs1, bs2;
    bs0[0] = sfb[voff];                 // row 0 / 2
    bs0[1] = sfb[voff + kHW];           // row 1 / 3
    bs1[0] = sfb[voff + 4 * kHW];       // row 4 / 6
    bs1[1] = sfb[voff + 5 * kHW];       // row 5 / 7
    bs2[0] = xr_v;                      // row 8 (x_rel) / row 10 (0)
    bs2[1] = yr_v;                      // row 9 (y_rel) / row 11 (0)

    // ---- Layer 1: h1 = relu(W1 @ feat + b1) ----
    v8f acc = {};
    acc = wmma_f32_k4(a1[0], bs0, acc);
    acc = wmma_f32_k4(a1[1], bs1, acc);
    acc = wmma_f32_k4(a1[2], bs2, acc);
    float h1[8];
#pragma unroll
    for (int r = 0; r < 8; ++r) h1[r] = fmaxf(acc[r] + bias1[r], 0.f);

    // ---- B operand for layer 2 via cndmask (duplication trick) ----
    v2f c0, c1;
    c0[0] = hi ? h1[2] : h1[0];
    c0[1] = hi ? h1[3] : h1[1];
    c1[0] = hi ? h1[6] : h1[4];
    c1[1] = hi ? h1[7] : h1[5];

    // ---- Layer 2: h2 = relu(W2 @ h1 + b2) ----
    v8f acc2 = {};
    acc2 = wmma_f32_k4(a2[0], c0, acc2);
    acc2 = wmma_f32_k4(a2[1], c1, acc2);

    // ---- Layer 3 + sigmoid (VALU; 1x8 dot) ----
    float z = b3;
#pragma unroll
    for (int r = 0; r < 8; ++r) z += w3v[r] * fmaxf(acc2[r] + bias2[r], 0.f);
    const float pred = 1.f / (1.f + __expf(-z));

    // ---- Dice partials: ALL lanes (hi lanes mirror lo lanes' pixel; same
    //      cachelines, so no extra traffic). Scaled by 0.5 after reduction.
    {
      const float t  = tg[tvoff];
      const float pm = pred * m;
      const float tm = t * m;
      inter += pm * tm;
      p2    += pm * pm;
      t2    += tm * tm;
    }

    voff  += kW;                        // next 128-pixel row of tiles
    tvoff += kW;
    yr_v  += yr_stp;
  }

  // Butterfly sum over all 32 lanes; every pixel counted twice -> x0.5.
#pragma unroll
  for (int off = 16; off >= 1; off >>= 1) {
    inter += __shfl_xor(inter, off, 32);
    p2    += __shfl_xor(p2,    off, 32);
    t2    += __shfl_xor(t2,    off, 32);
  }
  if (lane == 0) {
    red[wv][0] = inter * 0.5f;
    red[wv][1] = p2 * 0.5f;
    red[wv][2] = t2 * 0.5f;
  }
  __syncthreads();
  if (tid == 0) {
    float s0 = 0.f, s1 = 0.f, s2 = 0.f;
#pragma unroll
    for (int i = 0; i < 8; ++i) { s0 += red[i][0]; s1 += red[i][1]; s2 += red[i][2]; }
    partial[(size_t)blk * 3 + 0] = s0;
    partial[(size_t)blk * 3 + 1] = s1;
    partial[(size_t)blk * 3 + 2] = s2;
  }
}

__global__ __launch_bounds__(256) void seg_dice_finalize(
    const float* __restrict__ partial, float* __restrict__ out)
{
  __shared__ float lsh[8];
  const int tid  = threadIdx.x;
  const int lane = tid & 31;
  const int b    = tid >> 5;            // one wave per batch

  float inter = 0.f, p2 = 0.f, t2 = 0.f;
  for (int k = lane; k < kN; k += 32) {
    const float* p = partial + (size_t)(b * kN + k) * 3;
    inter += p[0]; p2 += p[1]; t2 += p[2];
  }
#pragma unroll
  for (int off = 16; off >= 1; off >>= 1) {
    inter += __shfl_xor(inter, off, 32);
    p2    += __shfl_xor(p2,    off, 32);
    t2    += __shfl_xor(t2,    off, 32);
  }
  if (lane == 0)
    lsh[b] = 1.f - (2.f * inter + 1.f) / (p2 + t2 + 1.f);
  __syncthreads();
  if (tid == 0) {
    float s = 0.f;
#pragma unroll
    for (int i = 0; i < 8; ++i) s += lsh[i];
    out[0] = s * 0.125f;
  }
}

extern "C" void kernel_launch(void* const* d_in, const int* in_sizes, int n_in,
                              void* d_out, int out_size, void* d_ws, size_t ws_size,
                              hipStream_t stream) {
  (void)in_sizes; (void)n_in; (void)out_size; (void)ws_size;
  const float* seg_feat    = (const float*)d_in[0];
  const float* conv_weight = (const float*)d_in[1];
  const float* mask        = (const float*)d_in[2];
  const int*   ind         = (const int*)d_in[3];
  const float* target      = (const float*)d_in[4];
  float* out     = (float*)d_out;
  float* partial = (float*)d_ws;        // 512*3 floats = 6 KB scratch

  seg_mlp_partial<<<kB * kN, 256, 0, stream>>>(seg_feat, conv_weight, mask, ind,
                                               target, partial);
  seg_dice_finalize<<<1, 256, 0, stream>>>(partial, out);
}